// PANConcDQL_54520314855765
// MI455X (gfx1250) — compile-verified
//
#include <hip/hip_runtime.h>
#include <cstdint>

#define BGR 16
#define NP 512
#define LSTEPS 20
#define CCH 3
#define NA 50
#define KKEEP 410
#define NNODES (BGR * NP)        // 8192
#define NF 13
#define MSZ (NP * NP)            // 262144
#define AMAT_TOT (BGR * MSZ)     // 4194304
#define EPS 1e-5f

typedef float v2f __attribute__((ext_vector_type(2)));
typedef float v8f __attribute__((ext_vector_type(8)));

// ---------------- wave helpers (wave32) ----------------
__device__ __forceinline__ float wave_sum(float v) {
#pragma unroll
  for (int o = 16; o > 0; o >>= 1) v += __shfl_xor(v, o, 32);
  return v;
}
__device__ __forceinline__ float wave_max(float v) {
#pragma unroll
  for (int o = 16; o > 0; o >>= 1) v = fmaxf(v, __shfl_xor(v, o, 32));
  return v;
}
__device__ __forceinline__ float lrelu(float x) { return x > 0.f ? x : 0.2f * x; }

// ---------------- chain init: A=0, T0=I, Macc=w0*I ----------------
__global__ void init_chain_kernel(float* __restrict__ Aadj, float* __restrict__ T0,
                                  float* __restrict__ Macc, const float* __restrict__ pan_w) {
  int e = blockIdx.x * blockDim.x + threadIdx.x;
  if (e >= AMAT_TOT) return;
  int ij = e & (MSZ - 1);
  float ident = ((ij >> 9) == (ij & 511)) ? 1.0f : 0.0f;
  Aadj[e] = 0.0f;
  T0[e] = ident;
  Macc[e] = pan_w[0] * ident;
}

// ---------------- scatter edges: A[g, dst%NP, src%NP] = 1 ----------------
__global__ void scatter_edges_kernel(const int* __restrict__ ei, float* __restrict__ Aadj, int E) {
  int e = blockIdx.x * blockDim.x + threadIdx.x;
  if (e >= E) return;
  int s = ei[e], d = ei[E + e];
  int g = d >> 9;
  Aadj[(size_t)g * MSZ + (size_t)(d & 511) * NP + (s & 511)] = 1.0f;
}

// ---------------- WMMA GEMM step: Tout = Tin @ A ; Macc += w[step]*Tout ----------------
// 2x2 register-blocked: one wave computes a 32x32 output block as four 16x16
// V_WMMA_F32_16X16X4_F32 tiles, so each A operand is reused across 2 B tiles
// and vice versa (4 WMMAs per 2 A-loads + 4 B-loads => ~8 flop/byte).
// A operand (16x4): lanes 0-15 hold M=0..15 with K={k,k+1}; lanes 16-31 K={k+2,k+3}.
// B operand (4x16): VGPR0 lanes 0-15 = row K=k, lanes 16-31 = row K=k+2; VGPR1 = K=k+1 / k+3.
// C/D: VGPR r -> M = r (+8 for upper half-wave), N = lane&15.
__global__ void __launch_bounds__(128)
pan_gemm_step_kernel(const float* __restrict__ Aadj, const float* __restrict__ Tin,
                     float* __restrict__ Tout, float* __restrict__ Macc,
                     const float* __restrict__ pan_w, int step) {
  const int wave = (blockIdx.x * blockDim.x + threadIdx.x) >> 5;
  const int lane = threadIdx.x & 31;
  const int b = wave >> 8;              // 16x16 blocks of 32x32 per graph
  const int tm = (wave >> 4) & 15;
  const int tn = wave & 15;
  const int row0 = tm << 5, col0 = tn << 5;
  const int half = lane >> 4;
  const int lr = lane & 15;

  const float* __restrict__ Ab = Aadj + (size_t)b * MSZ;
  const float* __restrict__ Tb = Tin + (size_t)b * MSZ;
  const float* arow0 = Tb + (size_t)(row0 + lr) * NP + (half << 1);
  const float* arow1 = arow0 + (size_t)16 * NP;
  const float* bcol0 = Ab + (size_t)(half << 1) * NP + col0 + lr;
  const float* bcol1 = bcol0 + 16;

  v8f c00 = {}, c01 = {}, c10 = {}, c11 = {};
#pragma unroll 2
  for (int k = 0; k < NP; k += 4) {
    v2f a0 = *(const v2f*)(arow0 + k);
    v2f a1 = *(const v2f*)(arow1 + k);
    v2f b0, b1;
    b0.x = bcol0[(size_t)k * NP];
    b0.y = bcol0[(size_t)(k + 1) * NP];
    b1.x = bcol1[(size_t)k * NP];
    b1.y = bcol1[(size_t)(k + 1) * NP];
    c00 = __builtin_amdgcn_wmma_f32_16x16x4_f32(false, a0, false, b0, (short)0, c00, false, false);
    c01 = __builtin_amdgcn_wmma_f32_16x16x4_f32(false, a0, false, b1, (short)0, c01, false, false);
    c10 = __builtin_amdgcn_wmma_f32_16x16x4_f32(false, a1, false, b0, (short)0, c10, false, false);
    c11 = __builtin_amdgcn_wmma_f32_16x16x4_f32(false, a1, false, b1, (short)0, c11, false, false);
  }

  const float w = pan_w[step];
  float* __restrict__ to = Tout + (size_t)b * MSZ;
  float* __restrict__ mo = Macc + (size_t)b * MSZ;
#pragma unroll
  for (int r = 0; r < 8; ++r) {
    int m0 = row0 + r + (half << 3);
    size_t o00 = (size_t)m0 * NP + col0 + lr;
    size_t o01 = o00 + 16;
    size_t o10 = o00 + (size_t)16 * NP;
    size_t o11 = o10 + 16;
    float v00 = c00[r], v01 = c01[r], v10 = c10[r], v11 = c11[r];
    to[o00] = v00; to[o01] = v01; to[o10] = v10; to[o11] = v11;
    mo[o00] += w * v00; mo[o01] += w * v01; mo[o10] += w * v10; mo[o11] += w * v11;
  }
}

// ---------------- deg -> dis = deg>0 ? rsqrt(deg) : 0 (wave per row) ----------------
__global__ void rowdis_kernel(const float* __restrict__ M, float* __restrict__ dis) {
  int wave = (blockIdx.x * blockDim.x + threadIdx.x) >> 5;
  int lane = threadIdx.x & 31;
  if (wave >= NNODES) return;
  const float* row = M + (size_t)wave * NP;
  float s = 0.f;
  for (int j = lane; j < NP; j += 32) s += row[j];
  s = wave_sum(s);
  if (lane == 0) dis[wave] = (s > 0.f) ? rsqrtf(s) : 0.f;
}

// ---------------- Mn = dis_i * M * dis_j (in place) ----------------
__global__ void normalize_kernel(float* __restrict__ M, const float* __restrict__ dis) {
  int e = blockIdx.x * blockDim.x + threadIdx.x;
  if (e >= AMAT_TOT) return;
  int b = e >> 18;
  int i = (e >> 9) & 511;
  int j = e & 511;
  M[e] *= dis[(b << 9) + i] * dis[(b << 9) + j];
}

// ---------------- colsum[b,n] = sum_i Mn[b,i,n] ----------------
__global__ void colsum_kernel(const float* __restrict__ M, float* __restrict__ cs) {
  int t = blockIdx.x * blockDim.x + threadIdx.x;
  if (t >= NNODES) return;
  int b = t >> 9, n = t & 511;
  const float* base = M + (size_t)b * MSZ + n;
  float s = 0.f;
  for (int i = 0; i < NP; ++i) s += base[(size_t)i * NP];
  cs[t] = s;
}

// ---------------- h = (Mn @ x) @ lin1_w + lin1_b   (wave per node) ----------------
__global__ void conv1_kernel(const float* __restrict__ Mn, const float* __restrict__ x,
                             const float* __restrict__ w, const float* __restrict__ bb,
                             float* __restrict__ h) {
  int wave = (blockIdx.x * blockDim.x + threadIdx.x) >> 5;
  int lane = threadIdx.x & 31;
  if (wave >= NNODES) return;
  int b = wave >> 9;
  const float* row = Mn + (size_t)wave * NP;
  float acc[NF];
#pragma unroll
  for (int f = 0; f < NF; ++f) acc[f] = 0.f;
  for (int j = lane; j < NP; j += 32) {
    float m = row[j];
    const float* xr = x + (size_t)((b << 9) + j) * NF;
#pragma unroll
    for (int f = 0; f < NF; ++f) acc[f] += m * xr[f];
  }
#pragma unroll
  for (int f = 0; f < NF; ++f) acc[f] = wave_sum(acc[f]);
  if (lane < CCH) {
    float s = bb[lane];
#pragma unroll
    for (int f = 0; f < NF; ++f) s += acc[f] * w[f * CCH + lane];
    h[(size_t)wave * CCH + lane] = s;
  }
}

// ---------------- GraphNorm (global, single block) + ReLU ----------------
__global__ void __launch_bounds__(1024)
graphnorm_relu_kernel(const float* __restrict__ in, float* __restrict__ out, int nrows,
                      const float* __restrict__ w, const float* __restrict__ bb,
                      const float* __restrict__ ms) {
  __shared__ float red[1024];
  __shared__ float stats[6];
  int tid = threadIdx.x;
  float s[3] = {0.f, 0.f, 0.f};
  for (int r = tid; r < nrows; r += blockDim.x) {
    s[0] += in[r * 3 + 0]; s[1] += in[r * 3 + 1]; s[2] += in[r * 3 + 2];
  }
  for (int c = 0; c < 3; ++c) {
    red[tid] = s[c]; __syncthreads();
    for (int st = (int)blockDim.x >> 1; st > 0; st >>= 1) {
      if (tid < st) red[tid] += red[tid + st];
      __syncthreads();
    }
    if (tid == 0) stats[c] = red[0] / (float)nrows;
    __syncthreads();
  }
  float mean[3] = {stats[0], stats[1], stats[2]};
  float m0 = ms[0], m1 = ms[1], m2 = ms[2];
  float v[3] = {0.f, 0.f, 0.f};
  for (int r = tid; r < nrows; r += blockDim.x) {
    float d0 = in[r * 3 + 0] - m0 * mean[0];
    float d1 = in[r * 3 + 1] - m1 * mean[1];
    float d2 = in[r * 3 + 2] - m2 * mean[2];
    v[0] += d0 * d0; v[1] += d1 * d1; v[2] += d2 * d2;
  }
  for (int c = 0; c < 3; ++c) {
    red[tid] = v[c]; __syncthreads();
    for (int st = (int)blockDim.x >> 1; st > 0; st >>= 1) {
      if (tid < st) red[tid] += red[tid + st];
      __syncthreads();
    }
    if (tid == 0) stats[3 + c] = red[0] / (float)nrows;
    __syncthreads();
  }
  float inv[3], wc[3], bc[3];
#pragma unroll
  for (int c = 0; c < 3; ++c) { inv[c] = rsqrtf(stats[3 + c] + EPS); wc[c] = w[c]; bc[c] = bb[c]; }
  float msv[3] = {m0, m1, m2};
  for (int r = tid; r < nrows; r += blockDim.x) {
#pragma unroll
    for (int c = 0; c < 3; ++c) {
      float d = in[r * 3 + c] - msv[c] * mean[c];
      out[r * 3 + c] = fmaxf(wc[c] * d * inv[c] + bc[c], 0.f);
    }
  }
}

// ---------------- pooling score ----------------
__global__ void score_kernel(const float* __restrict__ x1, const float* __restrict__ cs,
                             const float* __restrict__ p, const float* __restrict__ beta,
                             float* __restrict__ score) {
  int t = blockIdx.x * blockDim.x + threadIdx.x;
  if (t >= NNODES) return;
  const float* r = x1 + (size_t)t * 3;
  float d = r[0] * p[0] + r[1] * p[1] + r[2] * p[2];
  score[t] = tanhf(beta[0] * d + beta[1] * cs[t]);
}

// ---------------- per-graph top-K via bitonic sort (score desc, idx asc) ----------------
__global__ void __launch_bounds__(256)
topk_kernel(const float* __restrict__ score, const float* __restrict__ x1,
            int* __restrict__ idxK, float* __restrict__ xp) {
  __shared__ unsigned long long keys[NP];
  int b = blockIdx.x, tid = threadIdx.x;
  for (int i = tid; i < NP; i += blockDim.x) {
    unsigned u = __float_as_uint(score[b * NP + i]);
    u = (u & 0x80000000u) ? ~u : (u | 0x80000000u);   // order-preserving map
    keys[i] = (((unsigned long long)(~u)) << 32) | (unsigned)i;  // desc score, asc idx
  }
  __syncthreads();
  for (int k = 2; k <= NP; k <<= 1) {
    for (int j = k >> 1; j > 0; j >>= 1) {
      for (int i = tid; i < NP; i += blockDim.x) {
        int ixj = i ^ j;
        if (ixj > i) {
          bool up = ((i & k) == 0);
          unsigned long long a = keys[i], c = keys[ixj];
          if ((a > c) == up) { keys[i] = c; keys[ixj] = a; }
        }
      }
      __syncthreads();
    }
  }
  for (int kk = tid; kk < KKEEP; kk += blockDim.x) {
    int id = (int)(keys[kk] & 0xffffffffu);
    float v = score[b * NP + id];
    idxK[b * KKEEP + kk] = id;
#pragma unroll
    for (int c = 0; c < 3; ++c)
      xp[(size_t)(b * KKEEP + kk) * 3 + c] = x1[(size_t)(b * NP + id) * 3 + c] * v;
  }
}

// ---------------- xl/xr = xp @ W + b ----------------
__global__ void xlxr_kernel(const float* __restrict__ xp,
                            const float* __restrict__ wl, const float* __restrict__ bl,
                            const float* __restrict__ wr, const float* __restrict__ br,
                            float* __restrict__ xl, float* __restrict__ xr) {
  int t = blockIdx.x * blockDim.x + threadIdx.x;
  if (t >= BGR * KKEEP) return;
  float p0 = xp[t * 3 + 0], p1 = xp[t * 3 + 1], p2 = xp[t * 3 + 2];
#pragma unroll
  for (int c = 0; c < 3; ++c) {
    xl[t * 3 + c] = p0 * wl[c] + p1 * wl[3 + c] + p2 * wl[6 + c] + bl[c];
    xr[t * 3 + c] = p0 * wr[c] + p1 * wr[3 + c] + p2 * wr[6 + c] + br[c];
  }
}

// ---------------- GATv2 attention row: streaming softmax, wave per (b,i) ----------------
__global__ void __launch_bounds__(256)
gat_kernel(const float* __restrict__ Mn, const int* __restrict__ idxK,
           const float* __restrict__ xl, const float* __restrict__ xr,
           const float* __restrict__ att, const float* __restrict__ bias,
           float* __restrict__ g) {
  int wave = (blockIdx.x * blockDim.x + threadIdx.x) >> 5;
  int lane = threadIdx.x & 31;
  if (wave >= BGR * KKEEP) return;
  int b = wave / KKEEP, i = wave - b * KKEEP;
  const int* idxB = idxK + b * KKEEP;
  int idx_i = idxB[i];
  const float* MnB = Mn + (size_t)b * MSZ;
  const float* xlB = xl + (size_t)b * KKEEP * 3;
  float xr0 = xr[(size_t)wave * 3 + 0], xr1 = xr[(size_t)wave * 3 + 1], xr2 = xr[(size_t)wave * 3 + 2];
  float a0 = att[0], a1 = att[1], a2 = att[2];

  float mx = -3.0e38f;
  for (int j = lane; j < KKEEP; j += 32) {
    bool m = (j == i) || (MnB[(size_t)idxB[j] * NP + idx_i] != 0.0f);
    if (m) {
      float e0 = lrelu(xr0 + xlB[j * 3 + 0]);
      float e1 = lrelu(xr1 + xlB[j * 3 + 1]);
      float e2 = lrelu(xr2 + xlB[j * 3 + 2]);
      mx = fmaxf(mx, e0 * a0 + e1 * a1 + e2 * a2);
    }
  }
  mx = wave_max(mx);

  float sum = 0.f, n0 = 0.f, n1 = 0.f, n2 = 0.f;
  for (int j = lane; j < KKEEP; j += 32) {
    bool m = (j == i) || (MnB[(size_t)idxB[j] * NP + idx_i] != 0.0f);
    if (m) {
      float l0 = xlB[j * 3 + 0], l1 = xlB[j * 3 + 1], l2 = xlB[j * 3 + 2];
      float e0 = lrelu(xr0 + l0), e1 = lrelu(xr1 + l1), e2 = lrelu(xr2 + l2);
      float p = __expf(e0 * a0 + e1 * a1 + e2 * a2 - mx);
      sum += p; n0 += p * l0; n1 += p * l1; n2 += p * l2;
    }
  }
  sum = wave_sum(sum); n0 = wave_sum(n0); n1 = wave_sum(n1); n2 = wave_sum(n2);
  if (lane == 0) {
    float inv = 1.0f / sum;
    g[(size_t)wave * 3 + 0] = n0 * inv + bias[0];
    g[(size_t)wave * 3 + 1] = n1 * inv + bias[1];
    g[(size_t)wave * 3 + 2] = n2 * inv + bias[2];
  }
}

// ---------------- global add pool over K ----------------
__global__ void __launch_bounds__(256)
gp_kernel(const float* __restrict__ gn, float* __restrict__ gp) {
  __shared__ float red[256];
  int b = blockIdx.x, tid = threadIdx.x;
  float s[3] = {0.f, 0.f, 0.f};
  for (int k = tid; k < KKEEP; k += 256) {
    const float* r = gn + (size_t)(b * KKEEP + k) * 3;
    s[0] += r[0]; s[1] += r[1]; s[2] += r[2];
  }
  for (int c = 0; c < 3; ++c) {
    red[tid] = s[c]; __syncthreads();
    for (int st = 128; st > 0; st >>= 1) {
      if (tid < st) red[tid] += red[tid + st];
      __syncthreads();
    }
    if (tid == 0) gp[b * 3 + c] = red[0];
    __syncthreads();
  }
}

// ---------------- dueling head (one thread per graph) ----------------
__global__ void head_kernel(const float* __restrict__ x, const float* __restrict__ x1,
                            const float* __restrict__ gp, const int* __restrict__ cur,
                            const int* __restrict__ amask,
                            const float* __restrict__ v1w, const float* __restrict__ v1b,
                            const float* __restrict__ v2w, const float* __restrict__ v2b,
                            const float* __restrict__ v3w, const float* __restrict__ v3b,
                            const float* __restrict__ a1w, const float* __restrict__ a1b,
                            const float* __restrict__ a2w, const float* __restrict__ a2b,
                            const float* __restrict__ a3w, const float* __restrict__ a3b,
                            float* __restrict__ out) {
  int b = threadIdx.x;
  if (b >= BGR) return;
  int gidx = cur[b] + b * KKEEP;   // faithful to reference's pooled-offset indexing
  float feat[19];
#pragma unroll
  for (int f = 0; f < NF; ++f) feat[f] = x[(size_t)gidx * NF + f];
#pragma unroll
  for (int c = 0; c < 3; ++c) feat[13 + c] = x1[(size_t)gidx * 3 + c];
#pragma unroll
  for (int c = 0; c < 3; ++c) feat[16 + c] = gp[b * 3 + c];

  float h1[10], h2[5];
  for (int o = 0; o < 10; ++o) {
    float s = v1b[o];
    for (int i = 0; i < 19; ++i) s += feat[i] * v1w[i * 10 + o];
    h1[o] = fmaxf(s, 0.f);
  }
  for (int o = 0; o < 5; ++o) {
    float s = v2b[o];
    for (int i = 0; i < 10; ++i) s += h1[i] * v2w[i * 5 + o];
    h2[o] = fmaxf(s, 0.f);
  }
  float v = v3b[0];
  for (int i = 0; i < 5; ++i) v += h2[i] * v3w[i];

  float g1[10], g2[5], a3[NA];
  for (int o = 0; o < 10; ++o) {
    float s = a1b[o];
    for (int i = 0; i < 19; ++i) s += feat[i] * a1w[i * 10 + o];
    g1[o] = fmaxf(s, 0.f);
  }
  for (int o = 0; o < 5; ++o) {
    float s = a2b[o];
    for (int i = 0; i < 10; ++i) s += g1[i] * a2w[i * 5 + o];
    g2[o] = fmaxf(s, 0.f);
  }
  float amean = 0.f;
  for (int o = 0; o < NA; ++o) {
    float s = a3b[o];
    for (int i = 0; i < 5; ++i) s += g2[i] * a3w[i * NA + o];
    a3[o] = s; amean += s;
  }
  amean *= (1.0f / NA);
  for (int o = 0; o < NA; ++o) {
    float q = v + a3[o] - amean;
    out[b * NA + o] = (amask[b * NA + o] == 0) ? -1e8f : q;
  }
}

// ---------------- launch ----------------
extern "C" void kernel_launch(void* const* d_in, const int* in_sizes, int n_in,
                              void* d_out, int out_size, void* d_ws, size_t ws_size,
                              hipStream_t stream) {
  (void)n_in; (void)out_size; (void)ws_size;
  const float* x         = (const float*)d_in[0];
  const int*   edge_idx  = (const int*)d_in[1];
  const int*   cur       = (const int*)d_in[4];
  const int*   amask     = (const int*)d_in[5];
  const float* pan_w     = (const float*)d_in[6];
  const float* lin1_w    = (const float*)d_in[7];
  const float* lin1_b    = (const float*)d_in[8];
  const float* n1w       = (const float*)d_in[9];
  const float* n1b       = (const float*)d_in[10];
  const float* n1ms      = (const float*)d_in[11];
  const float* pool_p    = (const float*)d_in[12];
  const float* pool_beta = (const float*)d_in[13];
  const float* gwl       = (const float*)d_in[14];
  const float* gbl       = (const float*)d_in[15];
  const float* gwr       = (const float*)d_in[16];
  const float* gbr       = (const float*)d_in[17];
  const float* gatt      = (const float*)d_in[18];
  const float* gbias     = (const float*)d_in[19];
  const float* n2w       = (const float*)d_in[20];
  const float* n2b       = (const float*)d_in[21];
  const float* n2ms      = (const float*)d_in[22];
  const float* v1w = (const float*)d_in[23]; const float* v1b = (const float*)d_in[24];
  const float* v2w = (const float*)d_in[25]; const float* v2b = (const float*)d_in[26];
  const float* v3w = (const float*)d_in[27]; const float* v3b = (const float*)d_in[28];
  const float* a1w = (const float*)d_in[29]; const float* a1b = (const float*)d_in[30];
  const float* a2w = (const float*)d_in[31]; const float* a2b = (const float*)d_in[32];
  const float* a3w = (const float*)d_in[33]; const float* a3b = (const float*)d_in[34];
  const int E = in_sizes[1] / 2;

  float* ws = (float*)d_ws;
  float* A      = ws;
  float* T0     = A + AMAT_TOT;
  float* T1     = T0 + AMAT_TOT;
  float* Macc   = T1 + AMAT_TOT;         // becomes Mn in place
  float* dis    = Macc + AMAT_TOT;       // 8192
  float* colsum = dis + NNODES;          // 8192
  float* hbuf   = colsum + NNODES;       // 24576
  float* x1     = hbuf + NNODES * CCH;   // 24576
  float* scoreb = x1 + NNODES * CCH;     // 8192
  float* xp     = scoreb + NNODES;       // 19680
  float* xlb    = xp + BGR * KKEEP * 3;
  float* xrb    = xlb + BGR * KKEEP * 3;
  float* gb     = xrb + BGR * KKEEP * 3;
  float* gnb    = gb + BGR * KKEEP * 3;
  float* gpb    = gnb + BGR * KKEEP * 3; // 48
  int*   idxK   = (int*)(gpb + BGR * CCH);

  init_chain_kernel<<<AMAT_TOT / 256, 256, 0, stream>>>(A, T0, Macc, pan_w);
  scatter_edges_kernel<<<(E + 255) / 256, 256, 0, stream>>>(edge_idx, A, E);

  // 4096 waves (16 graphs x 16x16 blocks of 32x32), 4 waves/block
  float* tin = T0; float* tout = T1;
  for (int s = 1; s <= LSTEPS; ++s) {
    pan_gemm_step_kernel<<<BGR * 256 / 4, 128, 0, stream>>>(A, tin, tout, Macc, pan_w, s);
    float* t = tin; tin = tout; tout = t;
  }

  rowdis_kernel<<<NNODES * 32 / 256, 256, 0, stream>>>(Macc, dis);
  normalize_kernel<<<AMAT_TOT / 256, 256, 0, stream>>>(Macc, dis);
  colsum_kernel<<<NNODES / 256, 256, 0, stream>>>(Macc, colsum);
  conv1_kernel<<<NNODES * 32 / 256, 256, 0, stream>>>(Macc, x, lin1_w, lin1_b, hbuf);
  graphnorm_relu_kernel<<<1, 1024, 0, stream>>>(hbuf, x1, NNODES, n1w, n1b, n1ms);
  score_kernel<<<NNODES / 256, 256, 0, stream>>>(x1, colsum, pool_p, pool_beta, scoreb);
  topk_kernel<<<BGR, 256, 0, stream>>>(scoreb, x1, idxK, xp);
  xlxr_kernel<<<(BGR * KKEEP + 255) / 256, 256, 0, stream>>>(xp, gwl, gbl, gwr, gbr, xlb, xrb);
  gat_kernel<<<(BGR * KKEEP * 32 + 255) / 256, 256, 0, stream>>>(Macc, idxK, xlb, xrb, gatt, gbias, gb);
  graphnorm_relu_kernel<<<1, 1024, 0, stream>>>(gb, gnb, BGR * KKEEP, n2w, n2b, n2ms);
  gp_kernel<<<BGR, 256, 0, stream>>>(gnb, gpb);
  head_kernel<<<1, 32, 0, stream>>>(x, x1, gpb, cur, amask,
                                    v1w, v1b, v2w, v2b, v3w, v3b,
                                    a1w, a1b, a2w, a2b, a3w, a3b,
                                    (float*)d_out);
}